// loTeNet_28973849379437
// MI455X (gfx1250) — compile-verified
//
#include <hip/hip_runtime.h>
#include <hip/hip_bf16.h>
#include <cmath>

typedef __attribute__((ext_vector_type(2))) float v2f;
typedef __attribute__((ext_vector_type(8))) float v8f;

#define LD 36            // padded LDS row stride for G (floats)
#define DD 32            // bond dimension
#define WROW 72          // padded LDS row stride for raw weight block (floats)
                         // 2*WROW mod 64 = 16 -> both lane-halves conflict-free
#define WBUF (DD * WROW) // 2304 floats = 9216 bytes per weight buffer

// One block (128 threads = 4 waves) per patch.
// G = Mm_0 @ Mm_1 @ ... (32x32 chain, v_wmma_f32_16x16x4_f32), out = M0 @ (G @ vl).
// Weights stream HBM -> LDS via global_load_async_to_lds_b128 (double-buffered,
// ASYNCcnt-pipelined); the cos/sin contraction is fused into B-fragment reads.
__global__ __launch_bounds__(128) void lotenet_layer_kernel(
    const float* __restrict__ src,   // input image or previous layer output
    int is_image,                    // 1: src is 81x81x3 image, 0: src is (g*3,g*3,4)
    int pcols,                       // patches per row (27/9/3/1)
    int n,                           // features per patch (27 or 36)
    int out_dim,                     // output leg (4 or 10)
    const float* __restrict__ W0,    // [P,2,out,32]
    const float* __restrict__ Wm,    // [P,n-2,32,2,32]
    const float* __restrict__ Wl,    // [P,32,2]
    float* __restrict__ dst)         // [P,out]
{
    __shared__ float Ga[DD * LD];
    __shared__ float Gb[DD * LD];
    __shared__ float Wraw[2 * WBUF];     // double-buffered raw weight block
    __shared__ float ec[40], es[40];
    __shared__ float vls[DD], rs[DD];

    const int tid = threadIdx.x;
    const int p   = blockIdx.x;
    const int nm  = n - 2;                                   // # of Mm matrices
    const float* WmP = Wm + (size_t)p * nm * (DD * 2 * DD);  // per-patch base

    // ---- trig embedding of this patch's features (squeeze applied on the fly)
    if (tid < n) {
        const int chans = is_image ? 3 : 4;
        const int br = p / pcols, bc = p % pcols;
        const int ch = tid % chans;
        const int rest = tid / chans;
        const int dx = rest % 3, dy = rest / 3;
        const int Win = pcols * 3;
        float x = src[((br * 3 + dy) * Win + (bc * 3 + dx)) * chans + ch];
        float t = 0.5f * 3.14159265358979323846f * x;
        ec[tid] = cosf(t);
        es[tid] = sinf(t);
    }

    // ---- async HBM->LDS copy of one 8KB weight step, scattered so each
    //      256B row lands at 288B stride (bank-conflict-free B reads)
    const unsigned WrawOff = (unsigned)(uintptr_t)(void*)Wraw;
    auto issueAsync = [&](int step, int bufsel) {
        const char* gbase = (const char*)WmP + (size_t)step * (DD * 2 * DD * 4);
        const unsigned lbase = WrawOff + (unsigned)bufsel * (WBUF * 4);
#pragma unroll
        for (int j = 0; j < 4; ++j) {
            unsigned o = (unsigned)tid * 16u + (unsigned)j * 2048u;  // byte offset
            unsigned l = lbase + o + ((o >> 8) << 5);                // +32B pad/row
            const char* g = gbase + o;
            asm volatile("global_load_async_to_lds_b128 %0, %1, off"
                         :: "v"(l), "v"(g) : "memory");
        }
    };

    issueAsync(0, 0);
    if (nm > 1) {
        issueAsync(1, 1);
        asm volatile("s_wait_asynccnt 0x4" ::: "memory");
    } else {
        asm volatile("s_wait_asynccnt 0x0" ::: "memory");
    }
    __syncthreads();   // step-0 weights + embeddings visible to all waves

    // ---- G := Mm_0 = cos*W[:,0,:] + sin*W[:,1,:]
    {
        const float c = ec[1], s = es[1];
        const int d = tid >> 2, e0 = (tid & 3) * 8;
        const float* w = Wraw + d * WROW;
#pragma unroll
        for (int r = 0; r < 8; ++r)
            Ga[d * LD + e0 + r] = c * w[e0 + r] + s * w[32 + e0 + r];
    }
    __syncthreads();

    // ---- pipelined chain of 32x32x32 fp32 WMMA matmuls
    const int lane = tid & 31;
    const int wv   = tid >> 5;
    const int ti   = (wv >> 1) * 16;     // tile row base
    const int tj   = (wv & 1) * 16;      // tile col base
    const int half = lane >> 4;          // K sub-pair select
    const int lm   = lane & 15;          // M row / N col within tile
    const int nn   = tj + lm;

    float* Gsrc = Ga;
    float* Gdst = Gb;

    for (int i = 1; i < nm; ++i) {
        if (i + 1 < nm) {
            issueAsync(i + 1, (i + 1) & 1);              // prefetch next step
            asm volatile("s_wait_asynccnt 0x4" ::: "memory");   // step i landed
        } else {
            asm volatile("s_wait_asynccnt 0x0" ::: "memory");
        }
        __syncthreads();

        const float cs = ec[i + 1], ss = es[i + 1];
        const float* Wc = Wraw + (i & 1) * WBUF;

        v8f acc = {0.f, 0.f, 0.f, 0.f, 0.f, 0.f, 0.f, 0.f};
#pragma unroll
        for (int k = 0; k < 8; ++k) {
            const int kb = k * 4 + half * 2;
            v2f a, b;
            // A fragment: G rows ti..ti+15, K pair (kb, kb+1)
            a.x = Gsrc[(ti + lm) * LD + kb];
            a.y = Gsrc[(ti + lm) * LD + kb + 1];
            // B fragment: fused embed-contraction from raw weight block
            b.x = cs * Wc[kb * WROW + nn]       + ss * Wc[kb * WROW + 32 + nn];
            b.y = cs * Wc[(kb + 1) * WROW + nn] + ss * Wc[(kb + 1) * WROW + 32 + nn];
            acc = __builtin_amdgcn_wmma_f32_16x16x4_f32(
                false, a, false, b, (short)0, acc, false, false);
        }
        // D layout: VGPR v -> row (v + half*8) of the tile, col lm
#pragma unroll
        for (int v = 0; v < 8; ++v)
            Gdst[(ti + v + half * 8) * LD + tj + lm] = acc[v];
        __syncthreads();

        float* t2 = Gsrc; Gsrc = Gdst; Gdst = t2;
    }

    // ---- tail: vl contraction then M0 contraction (cheap VALU)
    if (tid < DD) {
        const float* wl = Wl + (size_t)p * DD * 2;
        vls[tid] = wl[tid * 2] * ec[n - 1] + wl[tid * 2 + 1] * es[n - 1];
    }
    __syncthreads();
    if (tid < DD) {
        float r = 0.f;
#pragma unroll 8
        for (int e = 0; e < DD; ++e) r += Gsrc[tid * LD + e] * vls[e];
        rs[tid] = r;
    }
    __syncthreads();
    if (tid < out_dim) {
        const float* w0a = W0 + ((size_t)p * 2 + 0) * out_dim * DD + tid * DD;
        const float* w0b = W0 + ((size_t)p * 2 + 1) * out_dim * DD + tid * DD;
        const float c0 = ec[0], s0 = es[0];
        float acc = 0.f;
#pragma unroll 8
        for (int d = 0; d < DD; ++d)
            acc += (c0 * w0a[d] + s0 * w0b[d]) * rs[d];
        dst[p * out_dim + tid] = acc;
    }
}

extern "C" void kernel_launch(void* const* d_in, const int* in_sizes, int n_in,
                              void* d_out, int out_size, void* d_ws, size_t ws_size,
                              hipStream_t stream) {
    const float* image = (const float*)d_in[0];
    const float* W0_0 = (const float*)d_in[1];
    const float* Wm_0 = (const float*)d_in[2];
    const float* Wl_0 = (const float*)d_in[3];
    const float* W0_1 = (const float*)d_in[4];
    const float* Wm_1 = (const float*)d_in[5];
    const float* Wl_1 = (const float*)d_in[6];
    const float* W0_2 = (const float*)d_in[7];
    const float* Wm_2 = (const float*)d_in[8];
    const float* Wl_2 = (const float*)d_in[9];
    const float* W0_3 = (const float*)d_in[10];
    const float* Wm_3 = (const float*)d_in[11];
    const float* Wl_3 = (const float*)d_in[12];

    float* v0 = (float*)d_ws;          // 729*4 floats
    float* v1 = v0 + 4096;             // 81*4 floats
    float* v2 = v1 + 1024;             // 9*4 floats
    float* out = (float*)d_out;        // 10 floats

    // Layer 0: image (81,81,3) -> 729 patches, n=27, out=4
    lotenet_layer_kernel<<<729, 128, 0, stream>>>(
        image, 1, 27, 27, 4, W0_0, Wm_0, Wl_0, v0);
    // Layer 1: v0 as (27,27,4) -> 81 patches, n=36, out=4
    lotenet_layer_kernel<<<81, 128, 0, stream>>>(
        v0, 0, 9, 36, 4, W0_1, Wm_1, Wl_1, v1);
    // Layer 2: v1 as (9,9,4) -> 9 patches, n=36, out=4
    lotenet_layer_kernel<<<9, 128, 0, stream>>>(
        v1, 0, 3, 36, 4, W0_2, Wm_2, Wl_2, v2);
    // Layer 3: v2 as (3,3,4) -> 1 patch, n=36, out=10
    lotenet_layer_kernel<<<1, 128, 0, stream>>>(
        v2, 0, 1, 36, 10, W0_3, Wm_3, Wl_3, out);
}